// IndicatorDistribution3_45741401703067
// MI455X (gfx1250) — compile-verified
//
#include <hip/hip_runtime.h>
#include <math.h>

// MI455X / gfx1250. Memory-bound elementwise kernel (0.7 FLOP/byte):
// optimal path = TDM bulk DMA to LDS + conflict-free LDS reads + coalesced
// NT streaming stores. No GEMM structure -> WMMA not applicable.

typedef __attribute__((ext_vector_type(4))) unsigned int v4u_t;
typedef __attribute__((ext_vector_type(8))) int          v8i_t;
typedef __attribute__((ext_vector_type(4))) int          v4i_t;

#define ROWS_PER_BLOCK 256
#define COLS 14

__global__ __launch_bounds__(256)
void IndicatorDistribution3_kernel(const float* __restrict__ state,
                                   float* __restrict__ out,
                                   int n_rows) {
    // Input tile: 256 rows x 14 f32 = 14336 B. Stride-14 dword reads are
    // bank-conflict-free in wave32 (gcd(14,64)=2 -> 32 distinct banks/wave).
    __shared__ float tile[ROWS_PER_BLOCK * COLS];
    // Output staging: stride-3 dword (gcd(3,64)=1 -> conflict-free), enables
    // fully coalesced unit-stride global stores.
    __shared__ float otile[ROWS_PER_BLOCK * 3];

    const int tid      = threadIdx.x;
    const int base_row = blockIdx.x * ROWS_PER_BLOCK;
    const int rows     = min(ROWS_PER_BLOCK, n_rows - base_row);
    const unsigned elems = (unsigned)(rows * COLS);

#if __has_builtin(__builtin_amdgcn_tensor_load_to_lds)
    // --- Tensor Data Mover: one DMA descriptor per block, issued by wave 0.
    // EXEC is ignored by TDM; branch is wave-uniform so exactly one issue.
    if (tid < 32) {
        unsigned long long gaddr =
            (unsigned long long)(const void*)(state + (size_t)base_row * COLS);
        unsigned lds_off = (unsigned)(unsigned long long)(const void*)&tile[0];

        // D# group 0: count=1, lds_addr, global_addr[56:0], type=2 ("image")
        v4u_t g0;
        g0.x = 1u;                                   // count=1, user mode
        g0.y = lds_off;                              // LDS byte address
        g0.z = (unsigned)gaddr;                      // global_addr[31:0]
        g0.w = ((unsigned)(gaddr >> 32) & 0x1FFFFFFu) | (2u << 30); // [56:32]|type

        // 1-D copy expressed as 2-D tile: tile_dim0 = tensor_dim0 = exact
        // element count (tail block shrinks tile -> no OOB reliance).
        const unsigned dim0 = elems, dim1 = 1u;
        const unsigned tile0 = elems, tile1 = 1u, tile2 = 0u;
        const unsigned stride0 = elems;              // irrelevant at dim1==1

        v8i_t g1;
        g1[0] = (int)(2u << 16);                                   // data_size=4B, wg_mask=0
        g1[1] = (int)((dim0 & 0xFFFFu) << 16);                     // tensor_dim0[15:0] @ bits63:48
        g1[2] = (int)(((dim0 >> 16) & 0xFFFFu) |                   // tensor_dim0[31:16]
                      ((dim1 & 0xFFFFu) << 16));                   // tensor_dim1[15:0]
        g1[3] = (int)(((dim1 >> 16) & 0xFFFFu) |                   // tensor_dim1[31:16]
                      ((tile0 & 0xFFFFu) << 16));                  // tile_dim0
        g1[4] = (int)((tile1 & 0xFFFFu) | ((tile2 & 0xFFFFu) << 16)); // tile_dim1, tile_dim2
        g1[5] = (int)stride0;                                      // tensor_dim0_stride[31:0]
        g1[6] = 0;                                                 // stride0[47:32], stride1[15:0]
        g1[7] = 0;                                                 // stride1[47:16]

        v4i_t g2 = {0, 0, 0, 0};           // no dims 2..4
        v4i_t g3 = {0, 0, 0, 0};
        v8i_t g4 = {0, 0, 0, 0, 0, 0, 0, 0};  // unused extra group (clang-23 6-arg form)

        __builtin_amdgcn_tensor_load_to_lds(g0, g1, g2, g3, g4, 0);
#if __has_builtin(__builtin_amdgcn_s_wait_tensorcnt)
        __builtin_amdgcn_s_wait_tensorcnt(0);
#else
        asm volatile("s_wait_tensorcnt 0x0" ::: "memory");
#endif
    }
#else
    // Fallback: coalesced cooperative copy (b32, unit stride across lanes).
    for (int j = tid; j < (int)elems; j += ROWS_PER_BLOCK)
        tile[j] = __builtin_nontemporal_load(&state[(size_t)base_row * COLS + j]);
#endif
    __syncthreads();   // publish LDS tile to all 8 waves

    if (tid < rows) {
        const float* r = &tile[tid * COLS];
        const float ha_open  = r[0];
        const float ha_close = r[1];
        const float lw       = r[5];
        const float uw       = r[6];
        const float ma       = r[8];
        const float srsi     = r[9];
        const float ssig     = r[10];
        const float bbu      = r[11];
        const float bbl      = r[13];
        // note: bb_middle (r[12]) only feeds squeeze&expansion, which is
        // identically false ((w<0.1)&(w>0.2), NaN-safe) -> exact zero term.

        const bool  bull  = ha_close > ha_open;
        const bool  bear  = ha_close < ha_open;
        const float body  = fabsf(ha_close - ha_open);
        const bool  sbull = bull && (body > 0.5f) && (uw < 1e-6f);
        const bool  sbear = bear && (body > 0.5f) && (lw < 1e-6f);

        const float pp    = (ha_close - bbl) / (bbu - bbl);
        const bool  pp_lo = pp < 0.2f;
        const bool  pp_hi = pp > 0.8f;
        const bool  ma_up = ma > 0.1f;
        const bool  ma_dn = ma < -0.1f;
        const bool  st_dn = ssig < -0.1f;
        const bool  st_up = ssig > 0.1f;

        const bool lsig = (sbull && ma_up && st_dn) || (sbull && ma_up && pp_lo) ||
                          (sbull && st_dn && pp_lo) || (ma_up && st_dn && pp_lo);
        const bool shsig = (sbear && ma_dn && st_up) || (sbear && ma_dn && pp_hi) ||
                           (sbear && st_up && pp_hi) || (ma_dn && st_up && pp_hi);

        const float col2 = 0.8f * (float)sbull + 0.7f * (float)ma_up +
                           0.8f * (float)(srsi < 0.2f) + 0.8f * (float)pp_lo +
                           1.8f * (float)lsig;
        const float col0 = 0.8f * (float)sbear + 0.7f * (float)ma_dn +
                           0.8f * (float)(srsi > 0.8f) + 0.8f * (float)pp_hi +
                           1.8f * (float)shsig;
        const float col1 = 1.5f;

        const float m   = fmaxf(fmaxf(col0, col2), col1);
        const float e0  = expf(col0 - m);
        const float e1  = expf(col1 - m);
        const float e2  = expf(col2 - m);
        const float inv = 1.0f / (e0 + e1 + e2);

        otile[tid * 3 + 0] = e0 * inv;
        otile[tid * 3 + 1] = e1 * inv;
        otile[tid * 3 + 2] = e2 * inv;
    }
    __syncthreads();   // repack complete -> coalesced streaming stores

    const long long obase = (long long)base_row * 3;
    const int lim = rows * 3;
#pragma unroll
    for (int i = 0; i < 3; ++i) {
        const int j = i * ROWS_PER_BLOCK + tid;
        if (j < lim)
            __builtin_nontemporal_store(otile[j], &out[obase + j]);
    }
}

extern "C" void kernel_launch(void* const* d_in, const int* in_sizes, int n_in,
                              void* d_out, int out_size, void* d_ws, size_t ws_size,
                              hipStream_t stream) {
    const float* state = (const float*)d_in[0];
    float*       out   = (float*)d_out;
    const int n_rows = in_sizes[0] / COLS;                 // 2,000,000
    const int blocks = (n_rows + ROWS_PER_BLOCK - 1) / ROWS_PER_BLOCK;
    IndicatorDistribution3_kernel<<<blocks, ROWS_PER_BLOCK, 0, stream>>>(state, out, n_rows);
}